// TaskLoRALinear_47193100648931
// MI455X (gfx1250) — compile-verified
//
#include <hip/hip_runtime.h>

typedef float v2f __attribute__((ext_vector_type(2)));
typedef float v8f __attribute__((ext_vector_type(8)));

#define IN_F   1024
#define OUT_F  1024
#define TASK_D 256
#define RANK   8
#define HID    64

// ---------------------------------------------------------------------------
// Kernel 1: h_{a,b}[t] = gelu(W1 . task_emb[t] + b1), for t in 0..3.
// 512 threads: tid = which*256 + t*64 + hid_idx
// ---------------------------------------------------------------------------
__global__ __launch_bounds__(512)
void gen_hidden_kernel(const float* __restrict__ task_emb,
                       const float* __restrict__ Wa1, const float* __restrict__ ba1,
                       const float* __restrict__ Wb1, const float* __restrict__ bb1,
                       float* __restrict__ h) {
  const int tid   = threadIdx.x;           // 0..511
  const int which = tid >> 8;              // 0 = gen_a, 1 = gen_b
  const int rem   = tid & 255;
  const int t     = rem >> 6;
  const int hidx  = rem & 63;
  const float* W    = which ? Wb1 : Wa1;
  const float* bias = which ? bb1 : ba1;
  const float* te = task_emb + t * TASK_D;
  const float* wr = W + hidx * TASK_D;
  float s = bias[hidx];
#pragma unroll 8
  for (int d = 0; d < TASK_D; ++d) s += wr[d] * te[d];
  // exact GELU: 0.5*x*(1+erf(x/sqrt(2)))
  float g = 0.5f * s * (1.0f + erff(s * 0.7071067811865476f));
  h[tid] = g;
}

// ---------------------------------------------------------------------------
// Kernel 2: vec_a[t] -> amat[t][1024][8]   (natural flat layout of vec_a)
//           vec_b[t] -> bmat[t][16][1024]  rows 0..7 = b_t, rows 8..15 = 0
// The zero padding rows make every lane's stage-1 B-fragment load valid,
// eliminating exec-mask divergence in the hot loop.
// 98304 threads: 65536 dot products + 32768 zero fills.
// ---------------------------------------------------------------------------
__global__ __launch_bounds__(256)
void gen_mats_kernel(const float* __restrict__ h,
                     const float* __restrict__ Wa2, const float* __restrict__ ba2,
                     const float* __restrict__ Wb2, const float* __restrict__ bb2,
                     float* __restrict__ amat, float* __restrict__ bmat) {
  const int tid = blockIdx.x * 256 + threadIdx.x;   // 0..98303
  if (tid >= 65536) {
    const int rem = tid - 65536;                    // 0..32767
    const int t   = rem >> 13;
    bmat[t * 16384 + 8192 + (rem & 8191)] = 0.0f;   // padding rows 8..15
    return;
  }
  const int which = tid >> 15;                      // 0 = a, 1 = b
  const int rem   = tid & 32767;
  const int t     = rem >> 13;                      // 0..3
  const int row   = rem & 8191;                     // 0..8191
  const float* W    = which ? Wb2 : Wa2;
  const float* bias = which ? bb2 : ba2;
  const float* hh = h + which * 256 + t * HID;
  const float* wr = W + (size_t)row * HID;
  float s = bias[row];
#pragma unroll
  for (int j = 0; j < HID; ++j) s += wr[j] * hh[j];
  if (which) bmat[t * 16384 + row] = s;             // vec_b flat == [r][k]
  else       amat[t * 8192  + row] = s;             // vec_a flat == [o][r]
}

// ---------------------------------------------------------------------------
// Kernel 3: main pass.  grid = (jt=128, t=4), block = 256 threads (8 waves).
// Each block: 16 rows {t + 4*(16*jt + m)}, all 1024 columns.
//   Stage 1 (per wave, k-eighth of 128): tmp16x8 += A(16x4,x)*B(4x16,padded b^T)
//   LDS reduce 8 partials (also does C-layout -> A-layout transpose)
//   Stage 2 (per wave, n-eighth of 128): C = base tile; C += tmp(16x8) * a_t^T
// x/base/out are streamed once -> non-temporal hints; amat/bmat stay hot in L2.
// 8 waves/block doubles machine-wide wave count (4096) for HBM latency hiding.
// ---------------------------------------------------------------------------
__global__ __launch_bounds__(256)
void lora_apply_kernel(const float* __restrict__ x, const float* __restrict__ base,
                       const float* __restrict__ amat, const float* __restrict__ bmat,
                       float* __restrict__ out) {
  const int t    = blockIdx.y;
  const int jt   = blockIdx.x;
  const int tid  = threadIdx.x;
  const int lane = tid & 31;
  const int w    = tid >> 5;      // wave id 0..7
  const int hs   = lane >> 4;     // half-wave select
  const int lp   = lane & 15;     // lane-in-half

  __shared__ float part[8][128];  // [wave][m*8 + r] stage-1 partial tmp
  __shared__ float tmpf[128];     // reduced tmp, m*8 + r

  const float* xrow = x + (size_t)(t + 4 * (jt * 16 + lp)) * IN_F;  // A row
  const float* at   = amat + t * (OUT_F * RANK);
  const float* brow = bmat + t * 16384 + lp * IN_F;  // rows 8..15 are zeros

  // ---- Stage 1: tmp = x_tile . b_t^T, wave w covers k in [w*128, w*128+128)
  v8f acc = {};
  const int kstart = w * 128 + hs * 2;     // A/B frag: K = base + hs*2 + {0,1}
#pragma unroll 8
  for (int kk = 0; kk < 128; kk += 4) {
    const int kb = kstart + kk;
    v2f af = __builtin_nontemporal_load((const v2f*)(xrow + kb));
    v2f bf = *(const v2f*)(brow + kb);
    acc = __builtin_amdgcn_wmma_f32_16x16x4_f32(
        false, af, false, bf, (short)0, acc, false, false);
  }
  // C layout: element (M = v + 8*hs, N = lp); keep only N < 8
  if (lp < RANK) {
#pragma unroll
    for (int v = 0; v < 8; ++v)
      part[w][(v + 8 * hs) * RANK + lp] = acc[v];
  }
  __syncthreads();
  if (tid < 128) {
    float s = part[0][tid];
#pragma unroll
    for (int p = 1; p < 8; ++p) s += part[p][tid];
    tmpf[tid] = s;
  }
  __syncthreads();

  // ---- Stage 2: y = base + tmp . a_t^T, wave w covers o in [w*128, w*128+128)
  // A fragments (16x8 tmp, two k=4 chunks), identical for all n-tiles:
  v2f a01 = *(const v2f*)(&tmpf[lp * RANK + hs * 2]);       // K = 0..3
  v2f a23 = *(const v2f*)(&tmpf[lp * RANK + 4 + hs * 2]);   // K = 4..7

#pragma unroll 2
  for (int it = 0; it < 8; ++it) {
    const int o = w * 128 + it * 16 + lp;                   // this lane's column
    // B fragment: B[k=r][n=o] = a_t[o][r]; lane needs (r, r+1) at r = hs*2 (+4)
    v2f b01 = *(const v2f*)(at + o * RANK + hs * 2);
    v2f b23 = *(const v2f*)(at + o * RANK + 4 + hs * 2);
    // C accumulator preloaded with base tile (SCALING = alpha/rank = 1.0)
    v8f cc;
#pragma unroll
    for (int v = 0; v < 8; ++v) {
      const size_t row = (size_t)(t + 4 * (jt * 16 + v + 8 * hs));
      cc[v] = __builtin_nontemporal_load(base + row * OUT_F + o);
    }
    cc = __builtin_amdgcn_wmma_f32_16x16x4_f32(
        false, a01, false, b01, (short)0, cc, false, false);
    cc = __builtin_amdgcn_wmma_f32_16x16x4_f32(
        false, a23, false, b23, (short)0, cc, false, false);
#pragma unroll
    for (int v = 0; v < 8; ++v) {
      const size_t row = (size_t)(t + 4 * (jt * 16 + v + 8 * hs));
      __builtin_nontemporal_store(cc[v], out + row * OUT_F + o);
    }
  }
}

// ---------------------------------------------------------------------------
extern "C" void kernel_launch(void* const* d_in, const int* in_sizes, int n_in,
                              void* d_out, int out_size, void* d_ws, size_t ws_size,
                              hipStream_t stream) {
  const float* x    = (const float*)d_in[0];
  const float* base = (const float*)d_in[1];
  const float* task = (const float*)d_in[2];
  const float* Wa1  = (const float*)d_in[3];
  const float* ba1  = (const float*)d_in[4];
  const float* Wa2  = (const float*)d_in[5];
  const float* ba2  = (const float*)d_in[6];
  const float* Wb1  = (const float*)d_in[7];
  const float* bb1  = (const float*)d_in[8];
  const float* Wb2  = (const float*)d_in[9];
  const float* bb2  = (const float*)d_in[10];
  float* out = (float*)d_out;

  // workspace: h[2][4][64] | amat[4][1024][8] | bmat[4][16][1024] (rows 8..15 = 0)
  float* h    = (float*)d_ws;      // 512 floats
  float* amat = h + 512;           // 32768 floats
  float* bmat = amat + 32768;      // 65536 floats

  gen_hidden_kernel<<<1, 512, 0, stream>>>(task, Wa1, ba1, Wb1, bb1, h);
  gen_mats_kernel<<<384, 256, 0, stream>>>(h, Wa2, ba2, Wb2, bb2, amat, bmat);

  dim3 grid(128, 4, 1);
  lora_apply_kernel<<<grid, 256, 0, stream>>>(x, base, amat, bmat, out);
}